// GlobalModel_67207648247828
// MI455X (gfx1250) — compile-verified
//
#include <hip/hip_runtime.h>
#include <hip/hip_bf16.h>

typedef float v2f __attribute__((ext_vector_type(2)));
typedef float v4f __attribute__((ext_vector_type(4)));
typedef float v8f __attribute__((ext_vector_type(8)));
typedef int   v4i __attribute__((ext_vector_type(4)));

#define ITERS_PER_LANE   32
#define NODES_PER_VEC    4
#define NODES_PER_WAVE   (ITERS_PER_LANE * NODES_PER_VEC * 32)   // 4096

// ---------------------------------------------------------------------------
// Kernel 0: zero the accumulators in workspace (sums[G*2], counts[G])
// ---------------------------------------------------------------------------
__global__ void zero_ws_kernel(float* __restrict__ ws, int n) {
    int i = blockIdx.x * blockDim.x + threadIdx.x;
    if (i < n) ws[i] = 0.0f;
}

// ---------------------------------------------------------------------------
// Kernel 1: segmented sum over sorted batch ids (bandwidth-bound, ~120MB).
// Wave w owns contiguous nodes [w*4096, (w+1)*4096). Lane l handles 4-node
// packets at node = base + (it*32 + l)*4: consecutive lanes -> consecutive
// 16B packets -> fully coalesced global_load_b128 (512B per wave instr).
// Each lane's id sequence is non-decreasing (batch is sorted), so we keep a
// running (sx,sy,cnt) and flush with 3 global f32 atomics only on id change.
// ---------------------------------------------------------------------------
__device__ __forceinline__ void flush_acc(float* __restrict__ sums,
                                          float* __restrict__ counts,
                                          int cur, float sx, float sy, float cnt) {
    atomicAdd(&sums[cur * 2 + 0], sx);
    atomicAdd(&sums[cur * 2 + 1], sy);
    atomicAdd(&counts[cur], cnt);
}

#define ACC_STEP(idv, vx, vy)                                              \
    do {                                                                   \
        if ((idv) != cur) {                                                \
            if (cur >= 0) flush_acc(sums, counts, cur, sx, sy, cnt);       \
            cur = (idv); sx = 0.0f; sy = 0.0f; cnt = 0.0f;                 \
        }                                                                  \
        sx += (vx); sy += (vy); cnt += 1.0f;                               \
    } while (0)

__global__ void seg_reduce_kernel(const float* __restrict__ x,
                                  const int* __restrict__ batch,
                                  float* __restrict__ sums,    // [G*2]
                                  float* __restrict__ counts,  // [G]
                                  int num_nodes) {
    int gtid = blockIdx.x * blockDim.x + threadIdx.x;
    int wave = gtid >> 5;
    int lane = gtid & 31;
    int base = wave * NODES_PER_WAVE;

    float sx = 0.0f, sy = 0.0f, cnt = 0.0f;
    int cur = -1;

    for (int it = 0; it < ITERS_PER_LANE; ++it) {
        int node = base + (it * 32 + lane) * NODES_PER_VEC;
        if (node >= num_nodes) break;               // monotonic per lane
        if (node + NODES_PER_VEC <= num_nodes) {
            // fast path: 3x b128 non-temporal loads (streamed exactly once)
            v4i ids = __builtin_nontemporal_load(
                reinterpret_cast<const v4i*>(batch + node));
            const v4f* xp = reinterpret_cast<const v4f*>(x) + (node >> 1);
            v4f xa = __builtin_nontemporal_load(xp);       // nodes node+0,+1
            v4f xb = __builtin_nontemporal_load(xp + 1);   // nodes node+2,+3
            ACC_STEP(ids.x, xa.x, xa.y);
            ACC_STEP(ids.y, xa.z, xa.w);
            ACC_STEP(ids.z, xb.x, xb.y);
            ACC_STEP(ids.w, xb.z, xb.w);
        } else {
            // tail: scalar per-node
            for (int j = 0; j < NODES_PER_VEC && node + j < num_nodes; ++j) {
                int id = batch[node + j];
                float2 v = reinterpret_cast<const float2*>(x)[node + j];
                ACC_STEP(id, v.x, v.y);
            }
        }
    }
    if (cur >= 0) flush_acc(sums, counts, cur, sx, sy, cnt);
}

// ---------------------------------------------------------------------------
// Kernel 2: per-tile MLP via V_WMMA_F32_16X16X4_F32.
//   out_row(g) = [u0, u1, mean0, mean1]          (A: 16x4, K=4)
//   h = relu(out @ W1^T + b1)  ;  y = h @ W2^T + b2
// B operand = W^T zero-padded to 4x16 (cols 4..15 zero).
// A layout: lane(0-15)->M, vgpr{0,1}=K{0,1}; lanes 16-31 carry K{2,3}.
// B layout: lane%16 -> N,  vgpr{0,1}=K{0,1} (lanes 0-15) / K{2,3} (16-31).
// D layout: lane%16 -> N,  vgpr v -> M = v + 8*(lane>=16).
// D->A conversion between the two GEMMs goes through a 16x4 LDS tile.
// One wave (32 threads) per block, one 16-graph tile per block.
// ---------------------------------------------------------------------------
__global__ void mlp_wmma_kernel(const float* __restrict__ u,
                                const float* __restrict__ sums,
                                const float* __restrict__ counts,
                                const float* __restrict__ W1,
                                const float* __restrict__ b1,
                                const float* __restrict__ W2,
                                const float* __restrict__ b2,
                                float* __restrict__ out, int G) {
    __shared__ float lds_h[16 * 4];

    const int lane = threadIdx.x;        // 0..31
    const int n    = lane & 15;          // N / M index within tile
    const int hi   = lane >> 4;          // 0: K=0,1   1: K=2,3
    const int tile = blockIdx.x;
    const int g    = min(tile * 16 + n, G - 1);

    // ---- A operand: [u0,u1,mean0,mean1] for graph g ----
    float cnt = counts[g];
    float inv = cnt > 0.0f ? 1.0f / cnt : 0.0f;   // where(cnt>0, sum/cnt, 0)
    v2f a;
    if (hi == 0) { a.x = u[g * 2 + 0];          a.y = u[g * 2 + 1]; }
    else         { a.x = sums[g * 2 + 0] * inv; a.y = sums[g * 2 + 1] * inv; }

    // ---- B operand: B[k][n] = W1[n][k] for n<4 else 0 ----
    v2f b;
    b.x = (n < 4) ? W1[n * 4 + 2 * hi + 0] : 0.0f;
    b.y = (n < 4) ? W1[n * 4 + 2 * hi + 1] : 0.0f;

    v8f c = {};
    v8f d = __builtin_amdgcn_wmma_f32_16x16x4_f32(
        false, a, false, b, (short)0, c, false, false);

    // ---- bias + relu, spill 16x4 h-tile to LDS in (M, K) order ----
    float bias1 = (n < 4) ? b1[n] : 0.0f;
    #pragma unroll
    for (int v = 0; v < 8; ++v) {
        int M = v + 8 * hi;
        float h = d[v] + bias1;
        h = h > 0.0f ? h : 0.0f;
        if (n < 4) lds_h[M * 4 + n] = h;
    }
    __syncthreads();

    // ---- second GEMM: A2 = h tile (reload in A layout), B2 = W2^T pad ----
    v2f a2;
    a2.x = lds_h[n * 4 + 2 * hi + 0];
    a2.y = lds_h[n * 4 + 2 * hi + 1];
    v2f b2v;
    b2v.x = (n < 4) ? W2[n * 4 + 2 * hi + 0] : 0.0f;
    b2v.y = (n < 4) ? W2[n * 4 + 2 * hi + 1] : 0.0f;

    v8f d2 = __builtin_amdgcn_wmma_f32_16x16x4_f32(
        false, a2, false, b2v, (short)0, c, false, false);

    float bias2 = (n < 4) ? b2[n] : 0.0f;
    #pragma unroll
    for (int v = 0; v < 8; ++v) {
        int M = v + 8 * hi;
        int row = tile * 16 + M;
        if (n < 4 && row < G) out[row * 4 + n] = d2[v] + bias2;
    }
}

// ---------------------------------------------------------------------------
// inputs: 0:x 1:edge_index 2:edge_attr 3:u 4:batch 5:W1 6:b1 7:W2 8:b2
// ---------------------------------------------------------------------------
extern "C" void kernel_launch(void* const* d_in, const int* in_sizes, int n_in,
                              void* d_out, int out_size, void* d_ws, size_t ws_size,
                              hipStream_t stream) {
    const float* x     = (const float*)d_in[0];
    const float* u     = (const float*)d_in[3];
    const int*   batch = (const int*)d_in[4];
    const float* W1    = (const float*)d_in[5];
    const float* b1    = (const float*)d_in[6];
    const float* W2    = (const float*)d_in[7];
    const float* b2    = (const float*)d_in[8];
    float*       out   = (float*)d_out;

    const int num_nodes = in_sizes[4];      // batch length
    const int G         = in_sizes[3] / 2;  // u is [G, 2]

    float* sums   = (float*)d_ws;           // [G*2]
    float* counts = sums + (size_t)G * 2;   // [G]

    // 1) zero accumulators (deterministic every call)
    {
        int n = G * 3;
        zero_ws_kernel<<<(n + 255) / 256, 256, 0, stream>>>(sums, n);
    }

    // 2) segmented reduction (streams ~120MB, bandwidth-bound)
    {
        int waves   = (num_nodes + NODES_PER_WAVE - 1) / NODES_PER_WAVE;
        int threads = 256;                  // 8 waves per block
        int blocks  = (waves * 32 + threads - 1) / threads;
        seg_reduce_kernel<<<blocks, threads, 0, stream>>>(x, batch, sums, counts,
                                                          num_nodes);
    }

    // 3) WMMA MLP over 16-graph tiles
    {
        int tiles = (G + 15) / 16;
        mlp_wmma_kernel<<<tiles, 32, 0, stream>>>(u, sums, counts, W1, b1, W2,
                                                  b2, out, G);
    }
}